// RoutingFeedforward_26422638805129
// MI455X (gfx1250) — compile-verified
//
#include <hip/hip_runtime.h>

// ---------------------------------------------------------------------------
// MoE routing feedforward for MI455X (gfx1250, wave32).
// fp32 end-to-end using V_WMMA_F32_16X16X4_F32 (exact fp32 matmul).
// Memory-bound problem (~85MB traffic @ 23.3 TB/s); dense-over-experts with
// per-tile expert skip keeps compute far below the bandwidth roofline while
// avoiding gather/scatter. Weight operand loads use uniform SGPR base +
// 32-bit lane index so the backend emits global_load ... s[base] scale_offset
// instead of per-load 64-bit VALU address adds.
// ---------------------------------------------------------------------------

typedef __attribute__((ext_vector_type(2))) float v2f;
typedef __attribute__((ext_vector_type(8))) float v8f;

#define T_TOKENS 4096   // B*S = 4*1024
#define DM 512          // d_model
#define DH 2048         // d_hidden
#define NE 8            // experts
#define MT 16           // tokens per workgroup tile (one 16-row WMMA M tile)
#define HC 128          // hidden chunk width (8 waves x 16 cols)
#define XROW (DM + 4)   // padded LDS row stride (bank-conflict avoidance)
#define HROW (HC + 4)

static __device__ __forceinline__ v8f wmma_f32(v2f a, v2f b, v8f c) {
  // 8 args: (neg_a, A, neg_b, B, c_mod, C, reuse_a, reuse_b)
  return __builtin_amdgcn_wmma_f32_16x16x4_f32(false, a, false, b, (short)0, c,
                                               false, false);
}

// ---------------------------------------------------------------------------
// Kernel 1: gating. One wave per token.
//   logits = x @ Wg^T + bg + K @ state ; probs = softmax ; gates = top2 mask
// ---------------------------------------------------------------------------
__global__ __launch_bounds__(256) void gating_kernel(
    const float* __restrict__ emb, const float* __restrict__ state,
    const float* __restrict__ Wg, const float* __restrict__ bg,
    const float* __restrict__ Kmat, float* __restrict__ probs_out,
    float* __restrict__ gates_out) {
  const int wave = threadIdx.x >> 5;
  const int lane = threadIdx.x & 31;
  const int t = blockIdx.x * 8 + wave;
  if (t >= T_TOKENS) return;

  const float* x = emb + (size_t)t * DM;
  float acc[NE];
#pragma unroll
  for (int e = 0; e < NE; ++e) acc[e] = 0.0f;

  for (int d = lane; d < DM; d += 32) {
    const float xd = x[d];
#pragma unroll
    for (int e = 0; e < NE; ++e) acc[e] = fmaf(xd, Wg[e * DM + d], acc[e]);
  }
  // wave32 reduction
#pragma unroll
  for (int e = 0; e < NE; ++e) {
#pragma unroll
    for (int off = 16; off > 0; off >>= 1) acc[e] += __shfl_xor(acc[e], off, 32);
  }

  if (lane == 0) {
    float logit[NE], p[NE];
#pragma unroll
    for (int e = 0; e < NE; ++e) {
      float tb = 0.0f;
#pragma unroll
      for (int f = 0; f < NE; ++f) tb = fmaf(Kmat[e * NE + f], state[t * NE + f], tb);
      logit[e] = acc[e] + bg[e] + tb;
    }
    float m = logit[0];
#pragma unroll
    for (int e = 1; e < NE; ++e) m = fmaxf(m, logit[e]);
    float s = 0.0f;
#pragma unroll
    for (int e = 0; e < NE; ++e) { p[e] = expf(logit[e] - m); s += p[e]; }
    const float inv = 1.0f / s;
#pragma unroll
    for (int e = 0; e < NE; ++e) p[e] *= inv;
    // top-2 (first occurrence on ties, matching lax.top_k)
    int i1 = 0;
#pragma unroll
    for (int e = 1; e < NE; ++e) if (p[e] > p[i1]) i1 = e;
    int i2 = (i1 == 0) ? 1 : 0;
#pragma unroll
    for (int e = 0; e < NE; ++e) if (e != i1 && p[e] > p[i2]) i2 = e;
#pragma unroll
    for (int e = 0; e < NE; ++e) {
      probs_out[t * NE + e] = p[e];
      gates_out[t * NE + e] = (e == i1 || e == i2) ? p[e] : 0.0f;
    }
  }
}

// ---------------------------------------------------------------------------
// Kernel 2: gated expert FFN. 16-token tile, 8 waves, WMMA f32 16x16x4.
//   out[t] = sum_e gate[t][e] * ( relu(x W1[e] + b1[e]) W2[e] + b2[e] )
// ---------------------------------------------------------------------------
__global__ __launch_bounds__(256) void moe_ffn_kernel(
    const float* __restrict__ emb, const float* __restrict__ gates,
    const float* __restrict__ W1, const float* __restrict__ b1,
    const float* __restrict__ W2, const float* __restrict__ b2,
    float* __restrict__ out) {
  __shared__ float sX[MT * XROW];   // x tile, fp32
  __shared__ float sH[MT * HROW];   // relu(x W1 + b1) chunk
  __shared__ float sG[MT * NE];     // gate values for this tile
  __shared__ float sFlag[NE];       // per-expert "tile uses expert" flag

  const int tid = threadIdx.x;
  const int wave = tid >> 5;
  const int lane = tid & 31;
  const int hf = lane >> 4;      // lane half: selects K pair / M half
  const int ln = lane & 15;      // N column (B/C/D) or M row (A)
  const int t0 = blockIdx.x * MT;

  // stage x tile + gates
  for (int i = tid; i < MT * DM; i += 256) {
    const int m = i >> 9, d = i & (DM - 1);
    sX[m * XROW + d] = emb[(size_t)(t0 + m) * DM + d];
  }
  for (int i = tid; i < MT * NE; i += 256) sG[i] = gates[(size_t)t0 * NE + i];
  __syncthreads();
  if (tid < NE) {
    float f = 0.0f;
    for (int m = 0; m < MT; ++m) f += sG[m * NE + tid];
    sFlag[tid] = f;
  }
  __syncthreads();

  // final output accumulators: wave owns N columns [wave*64, wave*64+64)
  v8f accOut[4];
#pragma unroll
  for (int nt = 0; nt < 4; ++nt) accOut[nt] = (v8f)0.0f;

  for (int e = 0; e < NE; ++e) {
    if (sFlag[e] == 0.0f) continue;  // uniform across block -> EXEC stays full

    const float* W1p = W1 + (size_t)e * DM * DH;   // uniform (SGPR) base
    const float* W2p = W2 + (size_t)e * DH * DM;   // uniform (SGPR) base

    v8f accO[4];
#pragma unroll
    for (int nt = 0; nt < 4; ++nt) accO[nt] = (v8f)0.0f;

    for (int ch = 0; ch < DH / HC; ++ch) {
      const int hbase = ch * HC;
      // uniform base for this chunk of W2; per-lane part stays a 32-bit index
      const float* W2c = W2p + (size_t)hbase * DM;
      // warm next chunk of W2 into cache (global_prefetch_b8)
      if (ch + 1 < DH / HC)
        __builtin_prefetch(&W2c[HC * DM + wave * 64 + ln], 0, 1);

      // ---- phase 1: h[:, hbase + wave*16 .. +16) = relu(x W1 + b1) -------
      {
        const int hcol = hbase + wave * 16 + ln;
        v8f hcc = (v8f)0.0f;
#pragma unroll 8
        for (int k0 = 0; k0 < DM; k0 += 4) {
          const int k = k0 + 2 * hf;
          // A (16x4 f32): VGPR0 = K0|K2 by lane half, VGPR1 = K1|K3
          v2f a = {sX[ln * XROW + k], sX[ln * XROW + k + 1]};
          // B (4x16 f32): lane half selects K pair, ln = N column.
          // 32-bit index from uniform base -> saddr + voffset scale_offset.
          const int bi = k * DH + hcol;
          v2f b = {W1p[bi], W1p[bi + DH]};
          hcc = wmma_f32(a, b, hcc);
        }
        const float bias = b1[e * DH + hcol];
#pragma unroll
        for (int r = 0; r < 8; ++r) {
          float v = hcc[r] + bias;
          v = v > 0.0f ? v : 0.0f;                        // relu
          sH[(r + 8 * hf) * HROW + wave * 16 + ln] = v;   // C: M = r + 8*half
        }
      }
      __syncthreads();

      // ---- phase 2: accO += h_chunk @ W2[hbase:hbase+HC, :] --------------
#pragma unroll 4
      for (int k0 = 0; k0 < HC; k0 += 4) {
        const int k = k0 + 2 * hf;
        v2f a = {sH[ln * HROW + k], sH[ln * HROW + k + 1]};
#pragma unroll
        for (int nt = 0; nt < 4; ++nt) {
          const int ncol = wave * 64 + nt * 16 + ln;
          const int bi = k * DM + ncol;   // 32-bit lane index, uniform base
          v2f b = {W2c[bi], W2c[bi + DM]};
          accO[nt] = wmma_f32(a, b, accO[nt]);
        }
      }
      __syncthreads();  // before next chunk overwrites sH
    }

    // ---- gate-weighted combine: accOut += g * (accO + b2) ----------------
#pragma unroll
    for (int nt = 0; nt < 4; ++nt) {
      const int ncol = wave * 64 + nt * 16 + ln;
      const float b2v = b2[e * DM + ncol];
#pragma unroll
      for (int r = 0; r < 8; ++r) {
        const float g = sG[(r + 8 * hf) * NE + e];
        accOut[nt][r] = fmaf(g, accO[nt][r] + b2v, accOut[nt][r]);
      }
    }
  }

  // write output tile
#pragma unroll
  for (int nt = 0; nt < 4; ++nt) {
    const int ncol = wave * 64 + nt * 16 + ln;
#pragma unroll
    for (int r = 0; r < 8; ++r) {
      const int m = r + 8 * hf;
      out[(size_t)(t0 + m) * DM + ncol] = accOut[nt][r];
    }
  }
}

// ---------------------------------------------------------------------------
extern "C" void kernel_launch(void* const* d_in, const int* in_sizes, int n_in,
                              void* d_out, int out_size, void* d_ws, size_t ws_size,
                              hipStream_t stream) {
  const float* emb   = (const float*)d_in[0];
  const float* state = (const float*)d_in[1];
  const float* Wg    = (const float*)d_in[2];
  const float* bg    = (const float*)d_in[3];
  const float* Kmat  = (const float*)d_in[4];
  const float* W1    = (const float*)d_in[5];
  const float* b1    = (const float*)d_in[6];
  const float* W2    = (const float*)d_in[7];
  const float* b2    = (const float*)d_in[8];

  float* out_main = (float*)d_out;                       // [T, DM]
  float* probs    = out_main + (size_t)T_TOKENS * DM;    // [T, NE]
  float* gates    = (float*)d_ws;                        // [T, NE] scratch

  gating_kernel<<<T_TOKENS / 8, 256, 0, stream>>>(emb, state, Wg, bg, Kmat,
                                                  probs, gates);
  moe_ffn_kernel<<<T_TOKENS / MT, 256, 0, stream>>>(emb, gates, W1, b1, W2, b2,
                                                    out_main);
}